// VerticalPooling_6837587935357
// MI455X (gfx1250) — compile-verified
//
#include <hip/hip_runtime.h>

// cummax along axis 0 of a [N=64, C=64, H=128, W=128] f32 tensor.
// Pure streaming scan: one thread owns one float4 "column" (4 adjacent
// (c,h,w) positions), walks the 64 planes keeping a running max in VGPRs.
//
// Memory-bound: 256 MiB in + 256 MiB out, each byte touched once.
// -> non-temporal TH hints on all global traffic (gfx1250 TH_NT), b128
//    vector loads/stores, 8-deep unroll for load pipelining.

typedef __attribute__((ext_vector_type(4))) float v4f;

__global__ __launch_bounds__(256) void cummax_axis0_kernel(
    const float* __restrict__ x, float* __restrict__ y,
    const int n_planes, const long long plane4 /* plane elems / 4 */)
{
    const long long i4 = (long long)blockIdx.x * 256 + threadIdx.x;
    if (i4 >= plane4) return;

    const v4f* __restrict__ xv = (const v4f*)x + i4;
    v4f* __restrict__ yv       = (v4f*)y + i4;

    // Plane 0: identity (cummax of first element is itself).
    v4f m = __builtin_nontemporal_load(xv);
    __builtin_nontemporal_store(m, yv);

    // Remaining planes: running max. Loads for the unrolled block are
    // address-independent, so the compiler issues them ahead of the
    // dependent v_max_num_f32 chain (only serial part, ~free vs HBM).
#pragma unroll 8
    for (int n = 1; n < n_planes; ++n) {
        v4f v = __builtin_nontemporal_load(xv + (long long)n * plane4);
        m.x = fmaxf(m.x, v.x);
        m.y = fmaxf(m.y, v.y);
        m.z = fmaxf(m.z, v.z);
        m.w = fmaxf(m.w, v.w);
        __builtin_nontemporal_store(m, yv + (long long)n * plane4);
    }
}

extern "C" void kernel_launch(void* const* d_in, const int* in_sizes, int n_in,
                              void* d_out, int out_size, void* d_ws, size_t ws_size,
                              hipStream_t stream) {
    const float* x = (const float*)d_in[0];
    float*       y = (float*)d_out;

    // Reference fixes shape [64, 64, 128, 128]; scan depth N = 64.
    const int       N      = 64;
    const long long total  = (long long)in_sizes[0];   // 67,108,864
    const long long plane  = total / N;                // 1,048,576 (divisible by 4)
    const long long plane4 = plane / 4;                // 262,144 float4 columns

    const int block = 256;                             // 8 wave32 waves
    const long long grid = (plane4 + block - 1) / block; // 1024 blocks, no tail

    cummax_axis0_kernel<<<(unsigned)grid, block, 0, stream>>>(x, y, N, plane4);
}